// GraphAttentionLayer_28114855919652
// MI455X (gfx1250) — compile-verified
//
#include <hip/hip_runtime.h>

typedef float v2f __attribute__((ext_vector_type(2)));
typedef float v8f __attribute__((ext_vector_type(8)));

#define B_   8
#define N_   1024
#define D_   256
#define H_   4
#define DH_  64

#define XSTR 260   // LDS row stride for X tile (260 % 64 == 4 -> conflict-free frag loads)
#define PSTR 1028  // LDS row stride for prob tile (1028 % 64 == 4)

__device__ __forceinline__ float leakyf(float x, float s) { return x > 0.f ? x : s * x; }

// ---------------------------------------------------------------------------
// Kernel A: h = X @ W^T   : (8192 x 256) = (8192 x 256) x (256 x 256)
// grid (128, 16), block 128 (4 waves). Each wave -> one 16x16 tile of a 64x16
// block tile. X tile + transposed W tile staged in LDS.
// ---------------------------------------------------------------------------
__global__ void gat_proj_wmma(const float* __restrict__ X, const float* __restrict__ W,
                              float* __restrict__ h)
{
    extern __shared__ float lds[];
    float* ldsX  = lds;               // 64 * XSTR floats
    float* ldsWt = lds + 64 * XSTR;   // 256 * 16 floats, Wt[k][n]
    const int tid  = threadIdx.x;
    const int wave = tid >> 5;
    const int lane = tid & 31;
    const int half = lane >> 4;
    const int l16  = lane & 15;
    const int m_block = blockIdx.x * 64;
    const int n0      = blockIdx.y * 16;

    // coalesced load of 64 rows of X
    for (int idx = tid; idx < 64 * 256; idx += 128) {
        int r = idx >> 8, c = idx & 255;
        ldsX[r * XSTR + c] = X[(size_t)(m_block + r) * D_ + c];
    }
    // coalesced load of 16 rows of W, stored transposed: Wt[k][n] = W[n0+n][k]
    for (int idx = tid; idx < 16 * 256; idx += 128) {
        int nl = idx >> 8, k = idx & 255;
        ldsWt[k * 16 + nl] = W[(size_t)(n0 + nl) * D_ + k];
    }
    __syncthreads();

    const int m0 = wave * 16;
    const float* arow = &ldsX[(m0 + l16) * XSTR];
    v8f acc = {};
#pragma unroll 4
    for (int k = 0; k < D_; k += 4) {
        const int kb = k + half * 2;           // ISA 16x4 f32 A layout: K = 2*(lane/16)+vgpr
        v2f af, bf;
        af.x = arow[kb];
        af.y = arow[kb + 1];
        bf.x = ldsWt[kb * 16 + l16];           // B[k][n] = W^T[k][n]
        bf.y = ldsWt[(kb + 1) * 16 + l16];
        acc = __builtin_amdgcn_wmma_f32_16x16x4_f32(false, af, false, bf,
                                                    (short)0, acc, false, false);
    }

    const int gcol = n0 + l16;
#pragma unroll
    for (int r = 0; r < 8; ++r) {
        const int gm = m_block + m0 + r + half * 8;   // C/D layout: M = r + 8*(lane/16)
        h[(size_t)gm * D_ + gcol] = acc[r];
    }
}

// ---------------------------------------------------------------------------
// Kernel B: s_left[b,i,h] = h[b,i,h,:] . wl ; s_right similarly with wr.
// One thread per (b, i, head). 32768 threads.
// ---------------------------------------------------------------------------
__global__ void gat_scores(const float* __restrict__ h, const float* __restrict__ aw,
                           float* __restrict__ sL, float* __restrict__ sR)
{
    const int g    = blockIdx.x * blockDim.x + threadIdx.x;  // = node*4 + head
    const int head = g & 3;
    const int node = g >> 2;                                  // b*N + i
    const float* hp = h + ((size_t)node * H_ + head) * DH_;
    float l = 0.f, r = 0.f;
#pragma unroll 8
    for (int k = 0; k < DH_; ++k) {
        const float v = hp[k];
        l += v * aw[k];
        r += v * aw[DH_ + k];
    }
    sL[g] = l;
    sR[g] = r;
}

// ---------------------------------------------------------------------------
// Kernel C: per (b, head, 16-row tile): masked-leaky scores -> softmax ->
// (16x1024) @ (1024x64) via WMMA -> fused residual+bias+leaky(0.5).
// block = 128 (4 waves), one wave per 16-col n-tile of the dh=64 output.
// Softmax normalization folded into the epilogue (1/rowsum scale).
// ---------------------------------------------------------------------------
__global__ void gat_attn_wmma(const float* __restrict__ X, const float* __restrict__ adj,
                              const float* __restrict__ h, const float* __restrict__ aw,
                              const float* __restrict__ sL, const float* __restrict__ sR,
                              const float* __restrict__ bias, float* __restrict__ out)
{
    extern __shared__ float P[];        // 16 * PSTR floats
    __shared__ float sLsh[16];
    __shared__ float rowscale[16];

    const int tid  = threadIdx.x;
    const int wave = tid >> 5;
    const int lane = tid & 31;
    const int half = lane >> 4;
    const int l16  = lane & 15;

    const int bx   = blockIdx.x;
    const int it   = bx & 63;
    const int head = (bx >> 6) & 3;
    const int b    = bx >> 8;
    const int i0   = it * 16;

    const float we = aw[2 * DH_];
    const float* adjRow = adj + ((size_t)b * N_ + i0) * N_;
    const float* sRb    = sR + (size_t)b * N_ * H_;

    if (tid < 16) sLsh[tid] = sL[((size_t)b * N_ + i0 + tid) * H_ + head];
    __syncthreads();

    // Stage 1: masked leaky scores into LDS (adjacency reads fully coalesced)
    for (int idx = tid; idx < 16 * N_; idx += 128) {
        const int r = idx >> 10, j = idx & (N_ - 1);
        const float a = adjRow[(size_t)r * N_ + j];
        float s = sLsh[r] + sRb[(size_t)j * H_ + head] + a * we;
        s = leakyf(s, 0.2f);
        if (a <= 1e-5f) s += -9.0e15f;
        P[r * PSTR + j] = s;
    }
    __syncthreads();

    // Stage 1b: each wave owns 4 rows: max, exp-in-place, sum (wave32 shfl reduce)
    for (int rr = 0; rr < 4; ++rr) {
        const int r = wave * 4 + rr;
        float* prow = &P[r * PSTR];
        float m = -3.4e38f;
        for (int j = lane; j < N_; j += 32) m = fmaxf(m, prow[j]);
        for (int off = 16; off; off >>= 1) m = fmaxf(m, __shfl_xor(m, off, 32));
        float sum = 0.f;
        for (int j = lane; j < N_; j += 32) {
            const float e = __expf(prow[j] - m);
            prow[j] = e;
            sum += e;
        }
        for (int off = 16; off; off >>= 1) sum += __shfl_xor(sum, off, 32);
        if (lane == 0) rowscale[r] = 1.0f / sum;
    }
    __syncthreads();

    // Stage 2: acc(16x16) = P(16x1024) @ h_head(1024x16-slice), K-chained WMMA
    const float* hHead = h + (size_t)b * N_ * D_ + head * DH_;
    const int n0 = wave * 16;
    const float* prow = &P[l16 * PSTR];
    v8f acc = {};
#pragma unroll 4
    for (int k = 0; k < N_; k += 4) {
        const int kb = k + half * 2;
        v2f af, bf;
        af.x = prow[kb];
        af.y = prow[kb + 1];
        bf.x = hHead[(size_t)kb * D_ + n0 + l16];
        bf.y = hHead[(size_t)(kb + 1) * D_ + n0 + l16];
        acc = __builtin_amdgcn_wmma_f32_16x16x4_f32(false, af, false, bf,
                                                    (short)0, acc, false, false);
    }

    // Epilogue: scale by 1/rowsum, residual + bias, leaky(0.5)
    const int n = n0 + l16;
    const int d = head * DH_ + n;
    const float bb = bias[d];
#pragma unroll
    for (int r = 0; r < 8; ++r) {
        const int gi = i0 + r + half * 8;
        const size_t o = ((size_t)b * N_ + gi) * D_ + d;
        const float v = X[o] + acc[r] * rowscale[r + half * 8] + bb;
        out[o] = leakyf(v, 0.5f);
    }
}

// ---------------------------------------------------------------------------
extern "C" void kernel_launch(void* const* d_in, const int* in_sizes, int n_in,
                              void* d_out, int out_size, void* d_ws, size_t ws_size,
                              hipStream_t stream)
{
    const float* X    = (const float*)d_in[0];  // (8,1024,256)
    const float* adj  = (const float*)d_in[1];  // (8,1024,1024)
    const float* W    = (const float*)d_in[2];  // (256,256)
    const float* aw   = (const float*)d_in[3];  // (1,129)
    const float* bias = (const float*)d_in[4];  // (256,)
    float* out = (float*)d_out;

    float* h  = (float*)d_ws;                     // 8192*256 floats
    float* sL = h + (size_t)B_ * N_ * D_;         // 32768 floats
    float* sR = sL + (size_t)B_ * N_ * H_;        // 32768 floats

    const size_t ldsA = (size_t)(64 * XSTR + 256 * 16) * sizeof(float);
    const size_t ldsC = (size_t)(16 * PSTR) * sizeof(float);

    gat_proj_wmma<<<dim3(128, 16), 128, ldsA, stream>>>(X, W, h);
    gat_scores<<<(B_ * N_ * H_) / 256, 256, 0, stream>>>(h, aw, sL, sR);
    gat_attn_wmma<<<B_ * H_ * (N_ / 16), 128, ldsC, stream>>>(X, adj, h, aw, sL, sR, bias, out);
}